// SpatialAttention_18683107738238
// MI455X (gfx1250) — compile-verified
//
#include <hip/hip_runtime.h>
#include <math.h>

// ---------------------------------------------------------------------------
// SpatialAttention (CBAM) for MI455X / gfx1250.
//
// Shapes: x [32,256,56,56] f32 (102.8 MB), w_conv [1,2,7,7] f32 (98 floats).
// Memory-bound (~0.4 FLOP/byte): plan = 1 HBM read of x + 1 HBM write of out;
// the second read of x (gate multiply) hits the 192MB L2. Conv is done as an
// im2col GEMM on v_wmma_f32_16x16x4_f32 (N=16 pixels/wave, K=98 taps).
//
// Pipeline (all on `stream`, dependencies via stream order):
//   K1a pool_partial : x -> per-slice (sum,max) partials   [8][32][2][3136]
//   K1b pool_combine : partials -> pooled (avg,max)        [32][2][3136]
//   K2  conv_gate    : pooled -(WMMA 7x7 conv)-> sigmoid -> gate [32][3136]
//   K3  apply_gate   : out = x * gate (float4, NT stores)
// Workspace: 6.4MB + 802KB + 401KB  (~7.7 MB)
// ---------------------------------------------------------------------------

typedef __attribute__((ext_vector_type(2))) float v2f;
typedef __attribute__((ext_vector_type(4))) float v4f;
typedef __attribute__((ext_vector_type(8))) float v8f;

#define B_      32
#define C_      256
#define H_      56
#define W_      56
#define HW      (H_ * W_)     // 3136
#define HW4     (HW / 4)      // 784  (float4 per image plane; 16B aligned rows)
#define NSLICE  8
#define CPS     (C_ / NSLICE) // 32 channels per slice

// ---------------- K1a: per-slice channel sum/max ---------------------------
__global__ __launch_bounds__(256)
void pool_partial(const float* __restrict__ x, float* __restrict__ partial) {
  int t   = blockIdx.x * 256 + threadIdx.x;     // [0, NSLICE*B_*HW4)
  int s   = t / (B_ * HW4);
  int r   = t - s * (B_ * HW4);
  int b   = r / HW4;
  int hw4 = r - b * HW4;                        // fastest-varying -> coalesced b128

  const v4f* xb = (const v4f*)x + (size_t)(b * C_ + s * CPS) * HW4 + hw4;

  v4f v   = xb[0];
  v4f sum = v;
  v4f mx  = v;
#pragma unroll 8
  for (int j = 1; j < CPS; ++j) {
    v4f u = xb[(size_t)j * HW4];
    sum += u;
    mx.x = fmaxf(mx.x, u.x);
    mx.y = fmaxf(mx.y, u.y);
    mx.z = fmaxf(mx.z, u.z);
    mx.w = fmaxf(mx.w, u.w);
  }
  v4f* p = (v4f*)partial;                       // [s][b][2][HW4]
  p[((size_t)(s * B_ + b) * 2 + 0) * HW4 + hw4] = sum;
  p[((size_t)(s * B_ + b) * 2 + 1) * HW4 + hw4] = mx;
}

// ---------------- K1b: combine slices -> pooled (avg, max) -----------------
__global__ __launch_bounds__(256)
void pool_combine(const float* __restrict__ partial, float* __restrict__ pooled) {
  int t   = blockIdx.x * 256 + threadIdx.x;     // [0, B_*HW4)
  int b   = t / HW4;
  int hw4 = t - b * HW4;

  const v4f* p = (const v4f*)partial;
  v4f sum = p[((size_t)(0 * B_ + b) * 2 + 0) * HW4 + hw4];
  v4f mx  = p[((size_t)(0 * B_ + b) * 2 + 1) * HW4 + hw4];
#pragma unroll
  for (int s = 1; s < NSLICE; ++s) {
    v4f su = p[((size_t)(s * B_ + b) * 2 + 0) * HW4 + hw4];
    v4f mu = p[((size_t)(s * B_ + b) * 2 + 1) * HW4 + hw4];
    sum += su;
    mx.x = fmaxf(mx.x, mu.x);
    mx.y = fmaxf(mx.y, mu.y);
    mx.z = fmaxf(mx.z, mu.z);
    mx.w = fmaxf(mx.w, mu.w);
  }
  v4f* q = (v4f*)pooled;                        // [b][2][HW4]: ch0=avg, ch1=max
  q[((size_t)b * 2 + 0) * HW4 + hw4] = sum * (1.0f / 256.0f);
  q[((size_t)b * 2 + 1) * HW4 + hw4] = mx;
}

// im2col fetch for tap k (k = ic*49 + kh*7 + kw), zero-padded, branch-free
// (clamped address + select keeps EXEC all-ones for the WMMA).
__device__ __forceinline__ float tapval(const float* __restrict__ pb,
                                        int h, int w, int k) {
  int  kk = (k < 98) ? k : 0;
  int  ic = kk / 49;
  int  r  = kk - ic * 49;
  int  kh = r / 7;
  int  kw = r - kh * 7;
  int  hh = h + kh - 3;
  int  ww = w + kw - 3;
  bool ok = (k < 98) && ((unsigned)hh < (unsigned)H_) && ((unsigned)ww < (unsigned)W_);
  int  hc = min(max(hh, 0), H_ - 1);
  int  wc = min(max(ww, 0), W_ - 1);
  float v = pb[ic * HW + hc * W_ + wc];
  return ok ? v : 0.0f;
}

// ---------------- K2: 7x7 conv via WMMA f32 16x16x4 + sigmoid --------------
// One block = one (b,h) row, 4 waves x 16 pixels (w 0..63; w>=56 masked on
// store only). A(16x4) = weights replicated over M rows: lanes 0-15 carry
// K=kb,kb+1 in VGPR0/1, lanes 16-31 carry K=kb+2,kb+3 (ISA f32 A layout).
// B(4x16) assumed symmetric: lane&15 = N (pixel), lane>>4 selects K pair.
// D row M=0 (acc[0], lanes 0-15) = conv(pixel N=lane); all rows identical.
__global__ __launch_bounds__(128)
void conv_gate(const float* __restrict__ pooled,
               const float* __restrict__ wconv,
               float* __restrict__ gate) {
  const int bh    = blockIdx.x;                 // [0, B_*H_)
  const int b     = bh / H_;
  const int h     = bh - b * H_;
  const int lane  = threadIdx.x & 31;
  const int wave  = threadIdx.x >> 5;           // 0..3
  const int n     = lane & 15;                  // pixel within 16-chunk
  const int khalf = (lane >> 4) << 1;           // 0 or 2
  const int w     = wave * 16 + n;              // output column (may be >= 56)

  const float* pb = pooled + (size_t)b * 2 * HW;

  v8f acc = {};
#pragma unroll
  for (int kb = 0; kb < 100; kb += 4) {         // 98 taps padded to 100
    int k0 = kb + khalf;
    float a0 = (k0     < 98) ? wconv[k0]     : 0.0f;  // broadcast (L1-hot)
    float a1 = (k0 + 1 < 98) ? wconv[k0 + 1] : 0.0f;
    float b0 = tapval(pb, h, w, k0);
    float b1 = tapval(pb, h, w, k0 + 1);
    v2f A = {a0, a1};
    v2f Bm = {b0, b1};
    acc = __builtin_amdgcn_wmma_f32_16x16x4_f32(
        /*neg_a=*/false, A, /*neg_b=*/false, Bm,
        /*c_mod=*/(short)0, acc, /*reuse_a=*/false, /*reuse_b=*/false);
  }

  float conv = acc[0];                          // M=0 row, N = lane (lanes 0-15)
  float g    = 1.0f / (1.0f + expf(-conv));
  if (lane < 16 && w < W_) {
    gate[(size_t)b * HW + h * W_ + w] = g;
  }
}

// ---------------- K3: out = x * gate (float4, NT stores) -------------------
__global__ __launch_bounds__(256)
void apply_gate(const float* __restrict__ x,
                const float* __restrict__ gate,
                float* __restrict__ out) {
  int i = blockIdx.x * 256 + threadIdx.x;       // [0, B_*C_*HW4)
  const v4f* x4 = (const v4f*)x;
  const v4f* g4 = (const v4f*)gate;             // [b][HW4], 4 consecutive w
  v4f xv = x4[i];                               // second read of x: L2-hot

  int bc  = i / HW4;                            // b*256 + c
  int rem = i - bc * HW4;                       // h*14 + w4
  int b   = bc >> 8;
  v4f g   = g4[(size_t)b * HW4 + rem];

  v4f o = xv * g;
  // Non-temporal store: dead output stream, keep x resident in L2.
  __builtin_nontemporal_store(o, (v4f*)out + i);
}

// ---------------------------------------------------------------------------
extern "C" void kernel_launch(void* const* d_in, const int* in_sizes, int n_in,
                              void* d_out, int out_size, void* d_ws, size_t ws_size,
                              hipStream_t stream) {
  const float* x     = (const float*)d_in[0];   // [32,256,56,56]
  const float* wconv = (const float*)d_in[1];   // [1,2,7,7] = 98 floats
  float*       out   = (float*)d_out;

  // Workspace carve-up (float offsets; all 16B aligned).
  float* partial = (float*)d_ws;                              // 8*32*2*3136 = 1,605,632 f
  float* pooled  = partial + (size_t)NSLICE * B_ * 2 * HW;    //   32*2*3136 =   200,704 f
  float* gate    = pooled  + (size_t)B_ * 2 * HW;             //     32*3136 =   100,352 f

  // K1a: 8*32*784 = 200,704 threads
  pool_partial<<<(NSLICE * B_ * HW4) / 256, 256, 0, stream>>>(x, partial);
  // K1b: 32*784 = 25,088 threads
  pool_combine<<<(B_ * HW4) / 256, 256, 0, stream>>>(partial, pooled);
  // K2: one block per (b,h) row; 4 waves x 16 pixels
  conv_gate<<<B_ * H_, 128, 0, stream>>>(pooled, wconv, gate);
  // K3: 32*256*784 = 6,422,528 float4s
  apply_gate<<<(B_ * C_ * HW4) / 256, 256, 0, stream>>>(x, gate, out);
  (void)in_sizes; (void)n_in; (void)out_size; (void)ws_size;
}